// SSIM3D_3143916060731
// MI455X (gfx1250) — compile-verified
//
#include <hip/hip_runtime.h>
#include <math.h>

typedef float v2f __attribute__((ext_vector_type(2)));
typedef float v8f __attribute__((ext_vector_type(8)));

#define VOL 160
#define TS 16            // output tile edge
#define PD 24            // padded tile edge (22 valid + 2 zero pad)
#define IXP 26           // x-stride of p/q tiles (even -> aligned ds_load_b64)
#define XP 17            // x-stride of t1/t2 (odd -> bank friendly)
#define SM_PQ  (PD*PD*IXP)          // 14976 floats per raw tile
#define SM_T1  (PD*PD*XP)           // 9792 floats
#define SM_T2  (PD*TS*XP)           // 6528 floats
#define SM_RED 8
#define SMEM_FLOATS (2*SM_PQ + SM_T1 + SM_T2 + SM_RED)   // 46280
#define SMEM_BYTES  (SMEM_FLOATS * 4)                    // 185120 B

__device__ __forceinline__ v8f wmma_f32x4(v2f a, v2f b, v8f c) {
  // fp32 WMMA: keeps E[x^2]-mu^2 cancellation numerically safe.
  return __builtin_amdgcn_wmma_f32_16x16x4_f32(
      false, a, false, b, (short)0, c, false, false);
}

__global__ void ssim3d_main(const float* __restrict__ pred,
                            const float* __restrict__ targ,
                            float* __restrict__ partial) {
  extern __shared__ float smem[];
  float* p_t = smem;                  // [PD z][PD y][IXP x], pads = 0
  float* q_t = smem + SM_PQ;
  float* t1  = smem + 2 * SM_PQ;      // [PD z][PD y][XP x]  (x-conv done)
  float* t2  = t1 + SM_T1;            // [PD z][TS y][XP x]  (y-conv done)
  float* red = t2 + SM_T2;            // 8 per-wave partials
  __shared__ float wsh[7];

  const int tid  = threadIdx.x;
  // force wave id into an SGPR so per-wave loops compile to scalar control flow
  const int wave = __builtin_amdgcn_readfirstlane(tid >> 5);
  const int lane = tid & 31;
  const int mi   = lane & 15;          // M (or N) index within half-wave
  const int hi   = (lane >> 4) & 1;    // half-wave select (K/M high bits)

  // Gaussian weights, sigma = 7/6 -> exp(-d^2 * 18/49), normalized.
  if (tid == 0) {
    float g[7]; float s = 0.f;
#pragma unroll
    for (int j = 0; j < 7; ++j) {
      float d = (float)(j - 3);
      g[j] = __expf(-d * d * (18.0f / 49.0f));
      s += g[j];
    }
#pragma unroll
    for (int j = 0; j < 7; ++j) wsh[j] = g[j] / s;
  }
  __syncthreads();

  // B fragments: banded 24x16 matrix B[k][n] = w[k-n], k = 4*slab + 2*hi + {0,1}.
  // Identical for every stage and field -> build once per thread.
  v2f bfr[6];
#pragma unroll
  for (int s5 = 0; s5 < 6; ++s5) {
    int k0 = 4 * s5 + 2 * hi;
    int d0 = k0 - mi, d1 = k0 + 1 - mi;
    bfr[s5].x = (d0 >= 0 && d0 < 7) ? wsh[d0] : 0.0f;
    bfr[s5].y = (d1 >= 0 && d1 < 7) ? wsh[d1] : 0.0f;
  }

  const int batch = blockIdx.z / 10;
  const int zt    = blockIdx.z % 10;
  const int bx0 = blockIdx.x * TS - 3;
  const int by0 = blockIdx.y * TS - 3;
  const int bz0 = zt * TS - 3;
  const size_t N3 = (size_t)VOL * VOL * VOL;
  const float* P = pred + (size_t)batch * N3;
  const float* Q = targ + (size_t)batch * N3;

  // ---- load haloed p/q tiles once; pads & out-of-volume -> 0 ----
  for (int idx = tid; idx < SM_PQ; idx += 256) {
    int x = idx % IXP;
    int rem = idx / IXP;
    int y = rem % PD;
    int z = rem / PD;
    int gx = bx0 + x, gy = by0 + y, gz = bz0 + z;
    float pv = 0.f, qv = 0.f;
    if (x < 22 && y < 22 && z < 22 && gx >= 0 && gx < VOL &&
        gy >= 0 && gy < VOL && gz >= 0 && gz < VOL) {
      size_t off = ((size_t)gz * VOL + gy) * VOL + gx;
      pv = P[off];
      qv = Q[off];
    }
    p_t[idx] = pv;
    q_t[idx] = qv;
  }
  __syncthreads();

  v8f res[5][2];   // stage-Z D fragments: [field][tile], live in VGPRs

#pragma unroll
  for (int f = 0; f < 5; ++f) {
    // ---- stage X: conv along x. 576 padded (z,y) lines -> 36 tiles, no guards.
    // Field values (p, q, p^2, q^2, p*q) are formed on the fly from ds_load_b64.
    for (int t = wave; t < 36; t += 8) {
      const int line = t * 16 + mi;
      const float* prow = p_t + line * IXP;
      const float* qrow = q_t + line * IXP;
      v2f a[6];
#pragma unroll
      for (int s5 = 0; s5 < 6; ++s5) {
        int k0 = 4 * s5 + 2 * hi;
        if (f == 0) {
          a[s5] = *(const v2f*)(prow + k0);
        } else if (f == 1) {
          a[s5] = *(const v2f*)(qrow + k0);
        } else if (f == 2) {
          v2f pv = *(const v2f*)(prow + k0); a[s5] = pv * pv;
        } else if (f == 3) {
          v2f qv = *(const v2f*)(qrow + k0); a[s5] = qv * qv;
        } else {
          v2f pv = *(const v2f*)(prow + k0);
          v2f qv = *(const v2f*)(qrow + k0); a[s5] = pv * qv;
        }
      }
      v8f c0 = {0.f,0.f,0.f,0.f,0.f,0.f,0.f,0.f};
      v8f c1 = {0.f,0.f,0.f,0.f,0.f,0.f,0.f,0.f};
      c0 = wmma_f32x4(a[0], bfr[0], c0);
      c1 = wmma_f32x4(a[1], bfr[1], c1);
      c0 = wmma_f32x4(a[2], bfr[2], c0);
      c1 = wmma_f32x4(a[3], bfr[3], c1);
      c0 = wmma_f32x4(a[4], bfr[4], c0);
      c1 = wmma_f32x4(a[5], bfr[5], c1);
      v8f c = c0 + c1;
      int line0 = t * 16 + hi * 8;
#pragma unroll
      for (int r = 0; r < 8; ++r) t1[(line0 + r) * XP + mi] = c[r];
    }
    __syncthreads();

    // ---- stage Y: conv along y. one tile per padded z (24 tiles), M = x ----
    for (int z = wave; z < PD; z += 8) {
      v2f a[6];
#pragma unroll
      for (int s5 = 0; s5 < 6; ++s5) {
        int k0 = 4 * s5 + 2 * hi;
        a[s5].x = t1[(z * PD + k0)     * XP + mi];
        a[s5].y = t1[(z * PD + k0 + 1) * XP + mi];
      }
      v8f c0 = {0.f,0.f,0.f,0.f,0.f,0.f,0.f,0.f};
      v8f c1 = {0.f,0.f,0.f,0.f,0.f,0.f,0.f,0.f};
      c0 = wmma_f32x4(a[0], bfr[0], c0);
      c1 = wmma_f32x4(a[1], bfr[1], c1);
      c0 = wmma_f32x4(a[2], bfr[2], c0);
      c1 = wmma_f32x4(a[3], bfr[3], c1);
      c0 = wmma_f32x4(a[4], bfr[4], c0);
      c1 = wmma_f32x4(a[5], bfr[5], c1);
      v8f c = c0 + c1;
#pragma unroll
      for (int r = 0; r < 8; ++r)
        t2[(z * TS + mi) * XP + hi * 8 + r] = c[r];
    }
    __syncthreads();

    // ---- stage Z: conv along z. 16 tiles (2/wave); D fragments stay in regs ----
#pragma unroll
    for (int ti = 0; ti < 2; ++ti) {
      int y = wave + 8 * ti;
      v2f a[6];
#pragma unroll
      for (int s5 = 0; s5 < 6; ++s5) {
        int k0 = 4 * s5 + 2 * hi;
        a[s5].x = t2[(k0       * TS + y) * XP + mi];
        a[s5].y = t2[((k0 + 1) * TS + y) * XP + mi];
      }
      v8f c0 = {0.f,0.f,0.f,0.f,0.f,0.f,0.f,0.f};
      v8f c1 = {0.f,0.f,0.f,0.f,0.f,0.f,0.f,0.f};
      c0 = wmma_f32x4(a[0], bfr[0], c0);
      c1 = wmma_f32x4(a[1], bfr[1], c1);
      c0 = wmma_f32x4(a[2], bfr[2], c0);
      c1 = wmma_f32x4(a[3], bfr[3], c1);
      c0 = wmma_f32x4(a[4], bfr[4], c0);
      c1 = wmma_f32x4(a[5], bfr[5], c1);
      res[f][ti] = c0 + c1;
    }
    // no barrier needed: next stage X touches only t1/p_t/q_t, not t2
  }

  // ---- fused SSIM map + local sum (all fragments share one layout) ----
  const float C1 = 1e-4f;     // 0.01^2
  const float C2 = 9e-4f;     // 0.03^2
  float sum = 0.f;
#pragma unroll
  for (int ti = 0; ti < 2; ++ti) {
#pragma unroll
    for (int r = 0; r < 8; ++r) {
      float mu1 = res[0][ti][r], mu2 = res[1][ti][r];
      float ex2 = res[2][ti][r], ey2 = res[3][ti][r], exy = res[4][ti][r];
      float m11 = mu1 * mu1, m22 = mu2 * mu2, m12 = mu1 * mu2;
      float s1 = ex2 - m11, s2 = ey2 - m22, s12 = exy - m12;
      float num = (2.f * m12 + C1) * (2.f * s12 + C2);
      float den = (m11 + m22 + C1) * (s1 + s2 + C2);
      sum += num / den;
    }
  }

  // deterministic reduction: wave32 butterfly, then fixed-order 8-way combine
#pragma unroll
  for (int off = 16; off > 0; off >>= 1) sum += __shfl_xor(sum, off, 32);
  if (lane == 0) red[wave] = sum;
  __syncthreads();
  if (tid == 0) {
    float t = 0.f;
#pragma unroll
    for (int i = 0; i < 8; ++i) t += red[i];
    int bid = blockIdx.x + 10 * (blockIdx.y + 10 * blockIdx.z);
    partial[bid] = t;
  }
}

__global__ void ssim3d_reduce(const float* __restrict__ partial,
                              float* __restrict__ out, int n) {
  __shared__ float s[256];
  float v = 0.f;
  for (int i = threadIdx.x; i < n; i += 256) v += partial[i];
  s[threadIdx.x] = v;
  __syncthreads();
  for (int st = 128; st > 0; st >>= 1) {
    if (threadIdx.x < st) s[threadIdx.x] += s[threadIdx.x + st];
    __syncthreads();
  }
  if (threadIdx.x == 0) {
    out[0] = 1.0f - s[0] * (1.0f / 8192000.0f);   // mean over 2*160^3
  }
}

extern "C" void kernel_launch(void* const* d_in, const int* in_sizes, int n_in,
                              void* d_out, int out_size, void* d_ws, size_t ws_size,
                              hipStream_t stream) {
  const float* pred = (const float*)d_in[0];
  const float* targ = (const float*)d_in[1];
  float* partial = (float*)d_ws;     // 2000 floats of block partials

  dim3 grid(10, 10, 20);             // 10x10x10 tiles x 2 batches (folded in z)
  ssim3d_main<<<grid, 256, SMEM_BYTES, stream>>>(pred, targ, partial);
  ssim3d_reduce<<<1, 256, 0, stream>>>(partial, (float*)d_out, 2000);
}